// ExpressionDWTLoss_83184926589628
// MI455X (gfx1250) — compile-verified
//
#include <hip/hip_runtime.h>

typedef __attribute__((ext_vector_type(2))) float v2f;
typedef __attribute__((ext_vector_type(4))) float v4f;
typedef __attribute__((ext_vector_type(8))) float v8f;

#define NSEQ 512
#define BW 50
#define BSTRIDE 112   // stored band width per row (101 needed + halo)
#define BHALO 56      // ldsD[i][j - i + BHALO]; valid cells use cols [6,106]
#define BIGF 1e8f

// Wrap-shuffle by -1 lane (lane l reads lane (l-1)&31) in pure VALU:
// DPP row_shr:1 within 16-lane rows, v_permlanex16 patches lanes 0 and 16
// (sel[0]=15 -> lane0<-lane31 (wrap), lane16<-lane15 (real neighbor)).
__device__ __forceinline__ float wrap_shr1(float v, bool row0) {
  int s = __builtin_bit_cast(int, v);
  int d = __builtin_amdgcn_update_dpp(s, s, 0x111, 0xf, 0xf, false);  // row_shr:1
  int f = __builtin_amdgcn_permlanex16(s, s, 0x0000000F, 0x00000000, false, false);
  return __builtin_bit_cast(float, row0 ? f : d);
}

// Wrap-shuffle by +1 lane (lane l reads lane (l+1)&31):
// DPP row_shl:1, permlanex16 with sel[15]=0 patches lanes 15 and 31
// (lane15<-lane16 (real neighbor), lane31<-lane0 (wrap)).
__device__ __forceinline__ float wrap_shl1(float v, bool row15) {
  int s = __builtin_bit_cast(int, v);
  int d = __builtin_amdgcn_update_dpp(s, s, 0x101, 0xf, 0xf, false);  // row_shl:1
  int f = __builtin_amdgcn_permlanex16(s, s, 0x00000000, 0x00000000, false, false);
  return __builtin_bit_cast(float, row15 ? f : d);
}

// One soft-DTW cell: softmin(gamma=2) of (va, vb, vc) + banded cost D[i][j].
// Cost load is unconditional with a clamped address; invalid cells emit BIG.
__device__ __forceinline__ float softdtw_cell(const float* __restrict__ ldsD,
                                              float va, float vb, float vc,
                                              int i, int j, int m, int p) {
  bool valid = (i >= 0) && (i < NSEQ) && (j >= 0) && (j < NSEQ) && (m <= BW);
  int ic = i < 0 ? 0 : (i > NSEQ - 1 ? NSEQ - 1 : i);
  int kc = 2 * m + p + (BHALO - BW);           // = (j-i)+BHALO when valid
  if (kc > BSTRIDE - 1) kc = BSTRIDE - 1;
  float dv = ldsD[ic * BSTRIDE + kc];          // induction-only addr: hides latency
  float mn = fminf(va, fminf(vb, vc));
  float ss = __expf((mn - va) * 0.5f) + __expf((mn - vb) * 0.5f) +
             __expf((mn - vc) * 0.5f);
  float rmin = mn - 2.0f * __logf(ss);         // -gamma*ln(sum exp(-(v-mn)/gamma))
  return valid ? (dv + rmin) : BIGF;
}

__global__ __launch_bounds__(128) void softdtw_band_wmma_kernel(
    const float* __restrict__ x, const float* __restrict__ y,
    float* __restrict__ ws)
{
  // 512*112 floats band costs + 32-float dump slot + norms = ~234 KB (<320KB/WGP)
  __shared__ float ldsD[NSEQ * BSTRIDE + 32];
  __shared__ float nA[NSEQ];
  __shared__ float nB[NSEQ];

  const int w = blockIdx.x;             // 0..31: xy, 32..63: xx, 64..95: yy
  const float* Am;
  const float* Bm;
  if (w < 32)      { Am = x + (size_t)w * (NSEQ * 4);        Bm = y + (size_t)w * (NSEQ * 4); }
  else if (w < 64) { Am = x + (size_t)(w - 32) * (NSEQ * 4); Bm = Am; }
  else             { Am = y + (size_t)(w - 64) * (NSEQ * 4); Bm = Am; }

  const int tid = threadIdx.x;

  // ---- Phase 0: squared norms (all 128 threads) ----
  for (int r = tid; r < NSEQ; r += 128) {
    v4f a = *(const v4f*)(Am + r * 4);
    nA[r] = a.x * a.x + a.y * a.y + a.z * a.z + a.w * a.w;
    v4f b = *(const v4f*)(Bm + r * 4);
    nB[r] = b.x * b.x + b.y * b.y + b.z * b.z + b.w * b.w;
  }
  __syncthreads();

  // ---- Phase 1: band cost matrix via V_WMMA_F32_16X16X4_F32 (K = D = 4) ----
  // D[i][j] = |a_i|^2 + |b_j|^2 - 2 a_i.b_j ; only |i-j| <= BHALO stored.
  const int wave = tid >> 5;
  const int lane = tid & 31;
  const int ln   = lane & 15;
  const int half = lane >> 4;
  v8f czero = {};
  int cnt = 0;
  for (int ti = 0; ti < 32; ++ti) {
    int tj0 = (ti - 4 < 0) ? 0 : ti - 4;
    int tj1 = (ti + 4 > 31) ? 31 : ti + 4;
    for (int tj = tj0; tj <= tj1; ++tj, ++cnt) {
      if ((cnt & 3) != wave) continue;   // uniform per wave: EXEC stays all-1s
      int arow = ti * 16 + ln;
      int bcol = tj * 16 + ln;
      // A 16x4 f32 layout: lane L holds a[L%16][2*(L/16)+v], v=0..1
      v2f av = *(const v2f*)(Am + arow * 4 + half * 2);
      v2f bv = *(const v2f*)(Bm + bcol * 4 + half * 2);
      bv = bv * -2.0f;                   // fold the -2 into B
      v8f acc = __builtin_amdgcn_wmma_f32_16x16x4_f32(
          false, av, false, bv, (short)0, czero, false, false);
      float nbv = nB[bcol];
      #pragma unroll
      for (int v = 0; v < 8; ++v) {
        int row = ti * 16 + v + 8 * half;       // C/D layout: M = v + 8*(L/16)
        int bc  = bcol - row + BHALO;           // N = L%16
        // Branchless: out-of-band elements go to a per-lane dump slot.
        bool inband = (unsigned)bc < (unsigned)BSTRIDE;
        int addr = inband ? (row * BSTRIDE + bc) : (NSEQ * BSTRIDE + lane);
        ldsD[addr] = nA[row] + nbv + acc[v];
      }
    }
  }
  __syncthreads();

  if (tid >= 32) return;   // DP runs on wave 0 only

  // ---- Phase 2: register-resident banded DP in band coordinates ----
  // k = j - i + 50 in [0,100]; active k have parity of d; slot m = (k-p)/2,
  // m = lane (slot 0) and lane+32 (slot 1). Neighbors:
  //   p=0: va = prev[m] (local),  vb = prev[m-1] (lane shift -1)
  //   p=1: va = prev[m+1] (+1),   vb = prev[m]   (local)
  //   vc = prevprev[m] (same-lane register, no comms)
  const int l = lane;
  const bool row0  = ((l & 15) == 0);
  const bool row15 = ((l & 15) == 15);
  float rp0 = BIGF, rp1 = BIGF;                 // diag d-1
  float rpp0 = (l == 25) ? 0.0f : BIGF;         // diag d-2; R[-1][-1]=0 at m=25
  float rpp1 = BIGF;

  for (int dh = 0; dh <= (NSEQ - 1); ++dh) {
    // even diagonal d = 2*dh (p = 0)
    {
      const int D2 = dh;
      float s0m1 = wrap_shr1(rp0, row0);        // == shfl(rp0, (l+31)&31)
      float s1m1 = wrap_shr1(rp1, row0);
      float vb0 = (l == 0) ? BIGF : s0m1;       // logical m-1; m=0 -> out of band
      float vb1 = (l == 0) ? s0m1 : s1m1;       // l==0: logical 31 = rp0 lane31
      float rn0 = softdtw_cell(ldsD, rp0, vb0, rpp0,
                               D2 + 25 - l, D2 + l - 25, l, 0);
      float rn1 = softdtw_cell(ldsD, rp1, vb1, rpp1,
                               D2 + 25 - (l + 32), D2 + (l + 32) - 25, l + 32, 0);
      rpp0 = rp0; rpp1 = rp1; rp0 = rn0; rp1 = rn1;
    }
    if (dh == NSEQ - 1) break;                  // d = 1022 was the last diagonal
    // odd diagonal d = 2*dh + 1 (p = 1)
    {
      const int D2 = dh;
      float s0p1 = wrap_shl1(rp0, row15);       // == shfl(rp0, (l+1)&31)
      float s1p1 = wrap_shl1(rp1, row15);
      float va0 = (l == 31) ? s1p1 : s0p1;      // l==31: logical 32 = rp1 lane0
      float va1 = s1p1;                         // l==31 wraps; that cell is invalid
      float rn0 = softdtw_cell(ldsD, va0, rp0, rpp0,
                               D2 + 25 - l, D2 + l + 1 - 25, l, 1);
      float rn1 = softdtw_cell(ldsD, va1, rp1, rpp1,
                               D2 + 25 - (l + 32), D2 + (l + 32) + 1 - 25, l + 32, 1);
      rpp0 = rp0; rpp1 = rp1; rp0 = rn0; rp1 = rn1;
    }
  }
  // Final R[N-1][M-1] lives at m=25 (k=50) on diagonal 1022, now in rp0.
  if (l == 25) ws[w] = rp0;
}

__global__ void softdtw_combine_kernel(const float* __restrict__ ws,
                                       float* __restrict__ out) {
  int t = threadIdx.x;
  if (t < 32) out[t] = ws[t] - 0.5f * (ws[32 + t] + ws[64 + t]);
}

extern "C" void kernel_launch(void* const* d_in, const int* in_sizes, int n_in,
                              void* d_out, int out_size, void* d_ws, size_t ws_size,
                              hipStream_t stream) {
  const float* x = (const float*)d_in[0];
  const float* y = (const float*)d_in[1];
  float* ws  = (float*)d_ws;   // 96 per-matrix soft-DTW scalars
  float* out = (float*)d_out;  // 32 floats
  softdtw_band_wmma_kernel<<<96, 128, 0, stream>>>(x, y, ws);
  softdtw_combine_kernel<<<1, 32, 0, stream>>>(ws, out);
}